// ReciprocalSpaceGatedAttention_72825465471208
// MI455X (gfx1250) — compile-verified
//
#include <hip/hip_runtime.h>
#include <hip/hip_bf16.h>
#include <math.h>

// Shapes (fixed by the reference): G=8, NG=256, H=128, M=128, N=2048.
#define G_   8
#define NG_  256
#define H_   128
#define M_   128
#define N_   (G_ * NG_)
#define TWO_PI_F 6.2831853071795864769f
#define SCALE_Q  0.08838834764831845f   // 1/sqrt(128)

typedef __attribute__((ext_vector_type(2))) float v2f;
typedef __attribute__((ext_vector_type(8))) float v8f;

// Native f32 WMMA: D(16x16) += A(16x4) * B(4x16).  A/B = 2 VGPRs/lane.
static __device__ __forceinline__ v8f wmma_f32(v2f a, v2f b, v8f c) {
  return __builtin_amdgcn_wmma_f32_16x16x4_f32(
      /*neg_a=*/false, a, /*neg_b=*/false, b,
      /*c_mod=*/(short)0, c, /*reuse_a=*/false, /*reuse_b=*/false);
}

static __device__ __forceinline__ float sigm(float x) {
  return 1.0f / (1.0f + expf(-x));
}

// Pair-interleaved B-side indexing: element (k, n) of a logical KxN matrix is
// stored at [((k>>1)*N + n)*2 + (k&1)] so rows k,k+1 at the same n are an
// adjacent 8-byte pair -> single coalesced global_load_b64 per WMMA B operand.

// ---------------------------------------------------------------------------
// Kernel 1: raw = node_feat @ [W_qkv ; W_vg]^T   (2048 x 512, K=128), f32 WMMA.
// 8 waves/block share one 16-row A strip staged in LDS; one 16x16 tile/wave.
// ---------------------------------------------------------------------------
__global__ __launch_bounds__(256) void k_qkv(const float* __restrict__ nf,
                                             const float* __restrict__ Wqkv,
                                             const float* __restrict__ Wvg,
                                             float* __restrict__ raw) {
  __shared__ float As[16 * 128];           // 8 KB: A strip shared by all waves
  const int wave = threadIdx.x >> 5, lane = threadIdx.x & 31;
  const int rowi = lane & 15, kb = (lane >> 4) << 1, hi = lane >> 4;
  const int tile = blockIdx.x * 8 + wave;
  const int r0 = (tile >> 5) * 16;         // same for all 8 waves in block
  const int c0 = (tile & 31) * 16;

  for (int i = threadIdx.x; i < 16 * 128; i += 256)
    As[i] = nf[(size_t)(r0 + (i >> 7)) * H_ + (i & 127)];
  __syncthreads();

  const int colg = c0 + rowi;
  const float* brow = (colg < 384) ? (Wqkv + (size_t)colg * H_)
                                   : (Wvg + (size_t)(colg - 384) * H_);
  v8f acc = {};
  for (int k0 = 0; k0 < H_; k0 += 4) {
    v2f a = *(const v2f*)(As + rowi * 128 + k0 + kb);       // ds_load_b64
    v2f b = *(const v2f*)(brow + k0 + kb);                  // global_load_b64
    acc = wmma_f32(a, b, acc);
  }
  for (int i = 0; i < 8; ++i)
    raw[(size_t)(r0 + i + 8 * hi) * 512 + c0 + rowi] = acc[i];
}

// ---------------------------------------------------------------------------
// Kernel 2: fractional coords  f = frac( (cell^T)^{-1} p ), Cramer per node.
// ---------------------------------------------------------------------------
__global__ __launch_bounds__(256) void k_frac(const float* __restrict__ pos,
                                              const float* __restrict__ cell,
                                              float* __restrict__ fw) {
  const int n = blockIdx.x * blockDim.x + threadIdx.x;
  const int g = n >> 8;
  const float* c = cell + g * 9;
  const float c00=c[0],c01=c[1],c02=c[2],c10=c[3],c11=c[4],c12=c[5],c20=c[6],c21=c[7],c22=c[8];
  const float cof00 =  (c11*c22 - c12*c21);
  const float cof01 = -(c10*c22 - c12*c20);
  const float cof02 =  (c10*c21 - c11*c20);
  const float cof10 = -(c01*c22 - c02*c21);
  const float cof11 =  (c00*c22 - c02*c20);
  const float cof12 = -(c00*c21 - c01*c20);
  const float cof20 =  (c01*c12 - c02*c11);
  const float cof21 = -(c00*c12 - c02*c10);
  const float cof22 =  (c00*c11 - c01*c10);
  const float inv = 1.0f / (c00*cof00 + c01*cof01 + c02*cof02);
  const float p0 = pos[n*3], p1 = pos[n*3+1], p2 = pos[n*3+2];
  float f0 = (cof00*p0 + cof01*p1 + cof02*p2) * inv;
  float f1 = (cof10*p0 + cof11*p1 + cof12*p2) * inv;
  float f2 = (cof20*p0 + cof21*p1 + cof22*p2) * inv;
  f0 -= floorf(f0); f1 -= floorf(f1); f2 -= floorf(f2);
  fw[n*3] = f0; fw[n*3+1] = f1; fw[n*3+2] = f2;
}

// ---------------------------------------------------------------------------
// Kernel 3: activations. q=silu, k=silu, v=v*sigmoid(vg+b).
// k / J*k stored transposed + pair-interleaved (over h); v pair-interleaved
// over n' (the K-dim of the final K*V GEMM).
// ---------------------------------------------------------------------------
__global__ __launch_bounds__(128) void k_act(const float* __restrict__ raw,
                                             const float* __restrict__ bvg,
                                             float* __restrict__ qs,
                                             float* __restrict__ vP,
                                             float* __restrict__ ktP,
                                             float* __restrict__ kjP) {
  const int n = blockIdx.x, h = threadIdx.x;
  const int g = n >> 8, ln = n & 255;
  const float* r = raw + (size_t)n * 512;
  float q = r[h];           q = q * sigm(q);
  float k = r[128 + h];     k = k * sigm(k);
  float v = r[256 + h] * sigm(r[384 + h] + bvg[h]);
  const int ho = h ^ 1;
  float ko = r[128 + ho];   ko = ko * sigm(ko);
  qs[(size_t)n * H_ + h] = q;
  const size_t kidx = (((size_t)g * 64 + (h >> 1)) * NG_ + ln) * 2 + (h & 1);
  ktP[kidx] = k;
  kjP[kidx] = (h & 1) ? -ko : ko;                    // (Jk)
  vP[(((size_t)g * 128 + (ln >> 1)) * H_ + h) * 2 + (ln & 1)] = v;
}

// ---------------------------------------------------------------------------
// Kernel 4: scalar head gates (tanh / sigmoid dot products).
// ---------------------------------------------------------------------------
__global__ __launch_bounds__(128) void k_gates(const float* __restrict__ nf,
                                               const float* __restrict__ Whg,
                                               const float* __restrict__ bhg,
                                               const float* __restrict__ Wmg,
                                               const float* __restrict__ bmg,
                                               float* __restrict__ hgs,
                                               float* __restrict__ out_gate) {
  const int n = blockIdx.x * blockDim.x + threadIdx.x;
  const float* x = nf + (size_t)n * H_;
  float ah = 0.f, am = 0.f;
  for (int h = 0; h < H_; ++h) { ah += x[h] * Whg[h]; am += x[h] * Wmg[h]; }
  hgs[n] = 1.0f + 0.2f * tanhf(ah + bhg[0]);
  out_gate[n] = sigm(am + bmg[0]);
}

// ---------------------------------------------------------------------------
// Kernel 5: phase matrices. CmP/SmP pair-interleaved over m (B operands);
// w-weighted CwT/SwT row-major over n (A operands).
// ---------------------------------------------------------------------------
__global__ __launch_bounds__(256) void k_phase(const float* __restrict__ fw,
                                               const int* __restrict__ nvec,
                                               const float* __restrict__ w,
                                               float* __restrict__ CmP,
                                               float* __restrict__ SmP,
                                               float* __restrict__ CwT,
                                               float* __restrict__ SwT) {
  const int gm = blockIdx.x;             // g*128 + m
  const int g = gm >> 7, m = gm & 127;
  const int n = threadIdx.x;
  const int node = g * NG_ + n;
  const float nx = (float)nvec[gm*3], ny = (float)nvec[gm*3+1], nz = (float)nvec[gm*3+2];
  const float ph = TWO_PI_F * (fw[node*3]*nx + fw[node*3+1]*ny + fw[node*3+2]*nz);
  float s, c;
  sincosf(ph, &s, &c);
  const float wv = w[gm];
  const size_t pidx = (((size_t)g * 64 + (m >> 1)) * NG_ + n) * 2 + (m & 1);
  CmP[pidx] = c;
  SmP[pidx] = s;
  CwT[((size_t)g * NG_ + n) * M_ + m] = wv * c;
  SwT[((size_t)g * NG_ + n) * M_ + m] = wv * s;
}

// ---------------------------------------------------------------------------
// Kernel 6: main. Per (g, 16-row strip): stage A strips (q, wC, wS) in LDS,
// WMMA-build K strip (16x256) in LDS, then upd = (1/256) K*V with gate
// epilogue. 4 waves/block; 128 blocks. All global B loads are paired b64.
// ---------------------------------------------------------------------------
__global__ __launch_bounds__(128) void k_main(const float* __restrict__ qs,
                                              const float* __restrict__ ktP,
                                              const float* __restrict__ kjP,
                                              const float* __restrict__ CwT,
                                              const float* __restrict__ SwT,
                                              const float* __restrict__ CmP,
                                              const float* __restrict__ SmP,
                                              const float* __restrict__ vP,
                                              const float* __restrict__ hgs,
                                              float* __restrict__ out) {
  __shared__ float qA[16 * 128];    // 8 KB  A strip: silu(q)
  __shared__ float cwA[16 * 128];   // 8 KB  A strip: w*cos
  __shared__ float swA[16 * 128];   // 8 KB  A strip: w*sin
  __shared__ float Kblk[16 * 256];  // 16 KB K strip
  const int g = blockIdx.x >> 4;
  const int rb = blockIdx.x & 15;
  const int gn0 = g * NG_ + rb * 16;       // global node row base
  const int wave = threadIdx.x >> 5, lane = threadIdx.x & 31;
  const int rowi = lane & 15, kb = (lane >> 4) << 1, hi = lane >> 4;

  // Stage the shared A strips once per block (coalesced b32 loads).
  for (int i = threadIdx.x; i < 16 * 128; i += 128) {
    const int r = i >> 7, k = i & 127;
    qA[i]  = qs [(size_t)(gn0 + r) * H_ + k];
    cwA[i] = CwT[(size_t)(gn0 + r) * M_ + k];
    swA[i] = SwT[(size_t)(gn0 + r) * M_ + k];
  }
  __syncthreads();

  // ---- Phase 1: K strip, 16 tiles of 16x16; 4 tiles per wave ----
  for (int t = wave * 4; t < wave * 4 + 4; ++t) {
    const int c0 = t * 16;
    const int colg = c0 + rowi;            // n' column for B lanes
    const float* bkt = ktP + ((size_t)g * 64 * NG_ + colg) * 2;
    const float* bkj = kjP + ((size_t)g * 64 * NG_ + colg) * 2;
    const float* bcm = CmP + ((size_t)g * 64 * NG_ + colg) * 2;
    const float* bsm = SmP + ((size_t)g * 64 * NG_ + colg) * 2;
    v8f c1 = {}, c2 = {}, wc = {}, wsa = {}, wsb = {};
    for (int k0 = 0; k0 < H_; k0 += 4) {       // C1 = q k^T, C2 = q (Jk)^T
      const int poff = ((k0 + kb) >> 1) * (NG_ * 2);
      v2f a  = *(const v2f*)(qA + rowi * 128 + k0 + kb);    // ds_load_b64
      v2f b1 = *(const v2f*)(bkt + poff);                   // global_load_b64
      v2f b2 = *(const v2f*)(bkj + poff);
      c1 = wmma_f32(a, b1, c1);
      c2 = wmma_f32(a, b2, c2);
    }
    for (int k0 = 0; k0 < M_; k0 += 4) {       // Wc, Ws over m (K=128)
      const int poff = ((k0 + kb) >> 1) * (NG_ * 2);
      v2f ac = *(const v2f*)(cwA + rowi * 128 + k0 + kb);
      v2f as = *(const v2f*)(swA + rowi * 128 + k0 + kb);
      v2f bc = *(const v2f*)(bcm + poff);
      v2f bs = *(const v2f*)(bsm + poff);
      wc  = wmma_f32(ac, bc, wc);
      wc  = wmma_f32(as, bs, wc);
      wsa = wmma_f32(as, bc, wsa);
      wsb = wmma_f32(ac, bs, wsb);
    }
    for (int i = 0; i < 8; ++i) {
      const int r = i + 8 * hi;
      Kblk[r * 256 + c0 + rowi] =
          SCALE_Q * (c1[i] * wc[i] + c2[i] * (wsa[i] - wsb[i]));
    }
  }
  __syncthreads();

  // ---- Phase 2: upd strip (16x128) = (1/256) K * V; 2 tiles per wave ----
  for (int u = wave * 2; u < wave * 2 + 2; ++u) {
    const int c0 = u * 16;
    const float* bv = vP + ((size_t)g * 128 * H_ + (c0 + rowi)) * 2;
    v8f acc = {};
    for (int k0 = 0; k0 < NG_; k0 += 4) {
      v2f a = *(const v2f*)(Kblk + rowi * 256 + k0 + kb);   // ds_load_b64
      v2f b = *(const v2f*)(bv + ((k0 + kb) >> 1) * (H_ * 2));
      acc = wmma_f32(a, b, acc);
    }
    for (int i = 0; i < 8; ++i) {
      const int ng = gn0 + i + 8 * hi;
      out[(size_t)ng * H_ + c0 + rowi] = acc[i] * (1.0f / 256.0f) * hgs[ng];
    }
  }
}

// ---------------------------------------------------------------------------
extern "C" void kernel_launch(void* const* d_in, const int* in_sizes, int n_in,
                              void* d_out, int out_size, void* d_ws, size_t ws_size,
                              hipStream_t stream) {
  (void)in_sizes; (void)n_in; (void)out_size; (void)ws_size;
  const float* nf   = (const float*)d_in[0];
  const float* pos  = (const float*)d_in[1];
  const float* cell = (const float*)d_in[2];
  const float* w    = (const float*)d_in[3];
  const float* Wqkv = (const float*)d_in[4];
  const float* Wvg  = (const float*)d_in[5];
  const float* bvg  = (const float*)d_in[6];
  const float* Whg  = (const float*)d_in[7];
  const float* bhg  = (const float*)d_in[8];
  const float* Wmg  = (const float*)d_in[9];
  const float* bmg  = (const float*)d_in[10];
  const int*   nvec = (const int*)d_in[11];

  float* ws  = (float*)d_ws;
  float* raw = ws;                         // 2048*512 = 1,048,576
  float* qs  = raw + (size_t)N_ * 512;     // 262,144 each below
  float* vP  = qs  + (size_t)N_ * H_;
  float* ktP = vP  + (size_t)N_ * H_;
  float* kjP = ktP + (size_t)N_ * H_;
  float* CwT = kjP + (size_t)N_ * H_;
  float* SwT = CwT + (size_t)N_ * H_;
  float* CmP = SwT + (size_t)N_ * H_;
  float* SmP = CmP + (size_t)G_ * M_ * NG_;
  float* fw  = SmP + (size_t)G_ * M_ * NG_;  // 6144
  float* hgs = fw  + (size_t)N_ * 3;         // 2048

  float* out      = (float*)d_out;
  float* out_gate = out + (size_t)N_ * H_;

  k_qkv  <<<512,  256, 0, stream>>>(nf, Wqkv, Wvg, raw);
  k_frac <<<8,    256, 0, stream>>>(pos, cell, fw);
  k_act  <<<2048, 128, 0, stream>>>(raw, bvg, qs, vP, ktP, kjP);
  k_gates<<<16,   128, 0, stream>>>(nf, Whg, bhg, Wmg, bmg, hgs, out_gate);
  k_phase<<<1024, 256, 0, stream>>>(fw, nvec, w, CmP, SmP, CwT, SwT);
  k_main <<<128,  128, 0, stream>>>(qs, ktP, kjP, CwT, SwT, CmP, SmP, vP, hgs, out);
}